// TemporalNMF_18511309046026
// MI455X (gfx1250) — compile-verified
//
#include <hip/hip_runtime.h>
#include <cstddef>

// D=32, B=512, N=2048, E=32, F=16, H=128, M=4
typedef __attribute__((ext_vector_type(2))) float v2f;
typedef __attribute__((ext_vector_type(8))) float v8f;

static constexpr size_t OFF_RE  = 33554432UL;  // rec_by_emb
static constexpr size_t OFF_FA  = 67108864UL;  // factors_by_age
static constexpr size_t OFF_FE  = 67371008UL;  // factors_by_emb
static constexpr size_t OFF_FO  = 67633152UL;  // factor_offsets
static constexpr size_t OFF_EMB = 67895296UL;  // embs

#define SACT 130   // activation/weight LDS stride: even (8B-aligned pairs), 130%64==2 -> conflict-free
#define SFAC 18    // factor LDS stride in k_rec: even, 18%64 -> distinct banks over 16 rows

__device__ __forceinline__ float leaky3(float x) { return x >= 0.0f ? x : 0.3f * x; }

__device__ __forceinline__ v8f wmma4(v2f a, v2f b, v8f c) {
  // V_WMMA_F32_16X16X4_F32 : D(16x16 f32) = A(16x4 f32) * B(4x16 f32) + C
  return __builtin_amdgcn_wmma_f32_16x16x4_f32(false, a, false, b, (short)0, c,
                                               false, false);
}

// ---------------------------------------------------------------- k_stats
// rstd = 1/std(ages, ddof=1) over all D*N = 65536 elements -> ws[0]
__global__ __launch_bounds__(256) void k_stats(const float* __restrict__ ages,
                                               float* __restrict__ ws) {
  __shared__ float sS[256];
  __shared__ float sQ[256];
  const int t = threadIdx.x;
  float s = 0.f, q = 0.f;
  for (int i = t; i < 65536; i += 256) {
    float x = ages[i];
    s += x;
    q += x * x;
  }
  sS[t] = s; sQ[t] = q;
  __syncthreads();
  for (int o = 128; o > 0; o >>= 1) {
    if (t < o) { sS[t] += sS[t + o]; sQ[t] += sQ[t + o]; }
    __syncthreads();
  }
  if (t == 0) {
    const float n = 65536.0f;
    const float mean = sS[0] / n;
    const float var = (sQ[0] - n * mean * mean) / (n - 1.0f);
    ws[0] = rsqrtf(var);
  }
}

// ---------------------------------------------------------------- k_om
// om = softplus(output_map); om = om / om.sum(axis=0) * F   -> ws[1..64]
__global__ __launch_bounds__(64) void k_om(const float* __restrict__ omap,
                                           float* __restrict__ ws) {
  __shared__ float sp[64];
  const int t = threadIdx.x;          // t = f*4 + m
  const float v = log1pf(expf(omap[t]));
  sp[t] = v;
  __syncthreads();
  const int m = t & 3;
  float sum = 0.f;
  for (int f = 0; f < 16; ++f) sum += sp[f * 4 + m];
  ws[1 + t] = v / sum * 16.0f;
}

// ---------------------------------------------------------------- k_embs
// embs = e * min(1, 1/(||e||+1e-7)), e = emb_table[idxs]  -> d_out[OFF_EMB..]
__global__ __launch_bounds__(128) void k_embs(const int* __restrict__ idxs,
                                              const float* __restrict__ etab,
                                              float* __restrict__ out_embs) {
  const int r = blockIdx.x * 128 + threadIdx.x;
  if (r >= 512) return;
  const float* e = etab + (size_t)idxs[r] * 32;
  float v[32];
  float ss = 0.f;
  for (int k = 0; k < 32; ++k) { v[k] = e[k]; ss += v[k] * v[k]; }
  const float sc = fminf(1.0f, 1.0f / (sqrtf(ss) + 1e-7f));
  for (int k = 0; k < 32; ++k) out_embs[(size_t)r * 32 + k] = v[k] * sc;
}

// ---------------------------------------------------------------- k_mlp
// 128 samples per block, 256 threads = 8 wave32. Activations 128x128 f32 in
// LDS (stride 130). Weights staged TRANSPOSED in LDS (sWT[n][k]) so every
// A and B fragment is one aligned ds_load_b64. Wave w owns tile-row w; its
// 16x128 A strip (32 x v2f) is register-cached and reused across all 8 tj.

__device__ __forceinline__ void gemm_hidden(const float* __restrict__ sIn,
                                            const float* __restrict__ sWT,
                                            const float* __restrict__ sBl,
                                            float* __restrict__ sOut,
                                            int lane, int wave) {
  const int mrow = lane & 15;
  const int kb   = ((lane >> 4) & 1) * 2;     // K base: lanes 16-31 -> +2
  const int hi   = ((lane >> 4) & 1) * 8;     // C rows: lanes 16-31 -> M+8
  const float* arow = sIn + (wave * 16 + mrow) * SACT + kb;
  v2f af[32];
#pragma unroll
  for (int g = 0; g < 32; ++g) af[g] = *(const v2f*)(arow + g * 4);
  for (int tj = 0; tj < 8; ++tj) {
    const int n = tj * 16 + mrow;
    const float* bcol = sWT + n * SACT + kb;
    v8f acc = {};
#pragma unroll
    for (int g = 0; g < 32; ++g)
      acc = wmma4(af[g], *(const v2f*)(bcol + g * 4), acc);
    const float bias = sBl[n];
#pragma unroll
    for (int v = 0; v < 8; ++v)
      sOut[(wave * 16 + v + hi) * SACT + n] = leaky3(acc[v] + bias);
  }
}

// 128 -> 16 output layer; W transposed in LDS (stride 130). Wave w -> tile-row w.
__device__ __forceinline__ v8f gemm_out16(const float* __restrict__ sIn,
                                          const float* __restrict__ sWoT,
                                          int lane, int wave) {
  const int mrow = lane & 15;
  const int kb   = ((lane >> 4) & 1) * 2;
  const float* arow = sIn + (wave * 16 + mrow) * SACT + kb;
  const float* bcol = sWoT + mrow * SACT + kb;
  v8f acc = {};
#pragma unroll
  for (int g = 0; g < 32; ++g)
    acc = wmma4(*(const v2f*)(arow + g * 4), *(const v2f*)(bcol + g * 4), acc);
  return acc;
}

__global__ __launch_bounds__(256) void k_mlp(
    const int* __restrict__ idxs, const float* __restrict__ ages,
    const float* __restrict__ ws,
    const float* __restrict__ a_w_in,  const float* __restrict__ a_b_in,
    const float* __restrict__ a_w_hid, const float* __restrict__ a_b_hid,
    const float* __restrict__ a_w_out, const float* __restrict__ a_b_out,
    const float* __restrict__ o_w_in,  const float* __restrict__ o_b_in,
    const float* __restrict__ o_w_hid, const float* __restrict__ o_b_hid,
    const float* __restrict__ o_w_out, const float* __restrict__ o_b_out,
    float* __restrict__ out) {
  __shared__ __align__(16) float sA[128 * SACT];
  __shared__ __align__(16) float sB[128 * SACT];
  __shared__ __align__(16) float sW[128 * SACT];   // transposed weights [n][k]
  __shared__ float sBias[128];
  __shared__ float sFA[128 * 17];   // factors_by_age tile (128 x 16)

  const int t    = threadIdx.x;
  const int lane = t & 31;
  const int wave = t >> 5;
  const int wgBase = blockIdx.x * 128;
  const float rstd = ws[0];

  const int row = t & 127;            // sample row within block
  const int hh  = (t >> 7) * 64;      // each half of threads does 64 features
  const int s   = wgBase + row;
  const int dd  = s >> 9;             // day
  const int bb  = s & 511;            // batch index
  const float age = fmaxf(ages[dd * 2048 + idxs[bb]] * rstd, 0.0f);

  // ---- MLP_a input layer (K=1): VALU
  for (int h = hh; h < hh + 64; ++h)
    sA[row * SACT + h] = leaky3(age * a_w_in[h] + a_b_in[h]);
  __syncthreads();

  float* pin = sA;
  float* pout = sB;
  for (int l = 0; l < 3; ++l) {
    const float* gw = a_w_hid + (size_t)l * 16384;
    for (int i = t; i < 16384; i += 256)
      sW[(i & 127) * SACT + (i >> 7)] = gw[i];              // transpose: [n][k]
    if (t < 128) sBias[t] = a_b_hid[l * 128 + t];
    __syncthreads();
    gemm_hidden(pin, sW, sBias, pout, lane, wave);
    __syncthreads();
    float* tmp = pin; pin = pout; pout = tmp;
  }
  // ---- MLP_a output layer (128 -> 16)
  for (int i = t; i < 2048; i += 256)
    sW[(i & 15) * SACT + (i >> 4)] = a_w_out[i];            // transpose: [n][k]
  __syncthreads();
  {
    const v8f facc = gemm_out16(pin, sW, lane, wave);
    const int hi = ((lane >> 4) & 1) * 8;
    const int f  = lane & 15;
    const float bo = a_b_out[f];
#pragma unroll
    for (int v = 0; v < 8; ++v) {
      const int r = wave * 16 + v + hi;
      const float val = facc[v] + bo;
      sFA[r * 17 + f] = val;
      out[OFF_FA + (size_t)(wgBase + r) * 16 + f] = val;
    }
  }
  __syncthreads();

  // ---- MLP_o input layer (K=33): VALU, embs read from d_out
  {
    float eb[32];
    const float* ep = out + OFF_EMB + (size_t)bb * 32;
#pragma unroll
    for (int k = 0; k < 32; ++k) eb[k] = ep[k];
    for (int h = hh; h < hh + 64; ++h) {
      float acc = o_b_in[h] + age * o_w_in[32 * 128 + h];
#pragma unroll
      for (int k = 0; k < 32; ++k) acc += eb[k] * o_w_in[k * 128 + h];
      sA[row * SACT + h] = leaky3(acc);
    }
  }
  __syncthreads();

  pin = sA; pout = sB;
  for (int l = 0; l < 4; ++l) {
    const float* gw = o_w_hid + (size_t)l * 16384;
    for (int i = t; i < 16384; i += 256)
      sW[(i & 127) * SACT + (i >> 7)] = gw[i];
    if (t < 128) sBias[t] = o_b_hid[l * 128 + t];
    __syncthreads();
    gemm_hidden(pin, sW, sBias, pout, lane, wave);
    __syncthreads();
    float* tmp = pin; pin = pout; pout = tmp;
  }
  // ---- MLP_o output layer; fuse factor_offsets + factors_by_emb stores
  for (int i = t; i < 2048; i += 256)
    sW[(i & 15) * SACT + (i >> 4)] = o_w_out[i];
  __syncthreads();
  {
    const v8f oacc = gemm_out16(pin, sW, lane, wave);
    const int hi = ((lane >> 4) & 1) * 8;
    const int f  = lane & 15;
    const float bo = o_b_out[f];
#pragma unroll
    for (int v = 0; v < 8; ++v) {
      const int r = wave * 16 + v + hi;
      const float off = oacc[v] + bo;
      const size_t gi = (size_t)(wgBase + r) * 16 + f;
      out[OFF_FO + gi] = off;
      out[OFF_FE + gi] = off + sFA[r * 17 + f];
    }
  }
}

// ---------------------------------------------------------------- k_rec
// Out[d,i,j,m] = sum_f relu(F[d,i,f]) * relu(F[d,j,f]) * om[f,m] + bias[d,m]
// = GEMM:  A(512x16) x B(16x2048), columns (j,m) contiguous in memory.
// Block = (d, rec, i-block of 128). 8 waves: wave w owns jm-tiles w,w+8,...
// A fragments for all 8 i-tiles register-cached (8x4 v2f); per-lane om
// coefficients register-cached; per jm-tile only 4 ds_load_b64 build B.
__global__ __launch_bounds__(256) void k_rec(const float* __restrict__ daily_bias,
                                             const float* __restrict__ ws,
                                             float* __restrict__ out) {
  __shared__ __align__(16) float sF[512 * SFAC];  // relu(factors), stride 18
  __shared__ float sOM[64];
  const int t    = threadIdx.x;
  const int lane = t & 31;
  const int wave = t >> 5;
  const int ib   = blockIdx.x & 3;
  const int rec  = (blockIdx.x >> 2) & 1;
  const int d    = blockIdx.x >> 3;

  const float* fac = out + (rec ? OFF_FE : OFF_FA) + (size_t)d * 8192;
  float* rbase = out + (rec ? OFF_RE : 0) + (size_t)d * 512 * 2048;

  for (int i = t; i < 8192; i += 256) {
    const int j = i >> 4, f = i & 15;
    sF[j * SFAC + f] = fmaxf(fac[i], 0.0f);
  }
  if (t < 64) sOM[t] = ws[1 + t];
  __syncthreads();

  const int nn = lane & 15;
  const int kb = ((lane >> 4) & 1) * 2;
  const int hi = ((lane >> 4) & 1) * 8;
  const int m  = lane & 3;                       // = ncol & 3 for any jm-tile
  const float dbias = daily_bias[d * 4 + m];

  // per-lane om coefficients for the two K slots of each k-group
  float omx[4], omy[4];
#pragma unroll
  for (int g = 0; g < 4; ++g) {
    omx[g] = sOM[(g * 4 + kb) * 4 + m];
    omy[g] = sOM[(g * 4 + kb + 1) * 4 + m];
  }
  // register-cache A fragments for this block's 8 i-tiles
  v2f af[8][4];
#pragma unroll
  for (int it = 0; it < 8; ++it) {
    const float* ar = sF + (ib * 128 + it * 16 + nn) * SFAC + kb;
#pragma unroll
    for (int g = 0; g < 4; ++g) af[it][g] = *(const v2f*)(ar + g * 4);
  }

  for (int jt = wave; jt < 128; jt += 8) {
    const int ncol = jt * 16 + nn;               // (j,m) flattened column
    const float* fr = sF + (ncol >> 2) * SFAC + kb;
    v2f bf[4];
#pragma unroll
    for (int g = 0; g < 4; ++g) {
      const v2f fj = *(const v2f*)(fr + g * 4);
      bf[g].x = fj.x * omx[g];
      bf[g].y = fj.y * omy[g];
    }
#pragma unroll
    for (int it = 0; it < 8; ++it) {
      v8f acc = {};
#pragma unroll
      for (int g = 0; g < 4; ++g) acc = wmma4(af[it][g], bf[g], acc);
      float* op = rbase + (size_t)(ib * 128 + it * 16 + hi) * 2048 + ncol;
#pragma unroll
      for (int v = 0; v < 8; ++v) op[(size_t)v * 2048] = acc[v] + dbias;
    }
  }
}

// ---------------------------------------------------------------- launch
extern "C" void kernel_launch(void* const* d_in, const int* in_sizes, int n_in,
                              void* d_out, int out_size, void* d_ws, size_t ws_size,
                              hipStream_t stream) {
  const int*   idxs    = (const int*)  d_in[0];
  const float* ages    = (const float*)d_in[1];
  const float* etab    = (const float*)d_in[2];
  const float* dbias   = (const float*)d_in[3];
  const float* omap    = (const float*)d_in[4];
  const float* a_w_in  = (const float*)d_in[5];
  const float* a_b_in  = (const float*)d_in[6];
  const float* a_w_hid = (const float*)d_in[7];
  const float* a_b_hid = (const float*)d_in[8];
  const float* a_w_out = (const float*)d_in[9];
  const float* a_b_out = (const float*)d_in[10];
  const float* o_w_in  = (const float*)d_in[11];
  const float* o_b_in  = (const float*)d_in[12];
  const float* o_w_hid = (const float*)d_in[13];
  const float* o_b_hid = (const float*)d_in[14];
  const float* o_w_out = (const float*)d_in[15];
  const float* o_b_out = (const float*)d_in[16];
  float* out = (float*)d_out;
  float* ws  = (float*)d_ws;

  k_stats<<<1, 256, 0, stream>>>(ages, ws);
  k_om<<<1, 64, 0, stream>>>(omap, ws);
  k_embs<<<4, 128, 0, stream>>>(idxs, etab, out + OFF_EMB);
  k_mlp<<<128, 256, 0, stream>>>(idxs, ages, ws,
                                 a_w_in, a_b_in, a_w_hid, a_b_hid, a_w_out, a_b_out,
                                 o_w_in, o_b_in, o_w_hid, o_b_hid, o_w_out, o_b_out,
                                 out);
  k_rec<<<256, 256, 0, stream>>>(dbias, ws, out);
}